// TTTLayer_11845519802374
// MI455X (gfx1250) — compile-verified
//
#include <hip/hip_runtime.h>

typedef __attribute__((ext_vector_type(2))) float f32x2;
typedef __attribute__((ext_vector_type(8))) float f32x8;

#define DEVI __device__ __forceinline__

constexpr int Bv = 8, Nv = 4096, Wv = 512, Hv = 8, Dv = 64;
constexpr float SCALE = 8.0f / (4096.0f * 512.0f);
constexpr float Nf = 4096.0f;

// workspace layout (float offsets)
constexpr size_t OFF_K    = 0;                                   // B*W*W
constexpr size_t OFF_XBAR = OFF_K + (size_t)Bv * Wv * Wv;        // B*W
constexpr size_t OFF_TRK  = OFF_XBAR + (size_t)Bv * Wv;          // B
constexpr size_t OFF_GG   = OFF_TRK + (size_t)Bv;                // H*D*D
constexpr size_t OFF_GP   = OFF_GG + (size_t)Hv * Dv * Dv;       // H*D
constexpr size_t OFF_KW   = OFF_GP + (size_t)Hv * Dv;            // B*H*2*W*D
constexpr size_t OFF_WEP  = OFF_KW + (size_t)Bv * Hv * 2 * Wv * Dv; // B*H*D*D
constexpr size_t OFF_BEP  = OFF_WEP + (size_t)Bv * Hv * Dv * Dv; // B*H*D
constexpr size_t OFF_SCR  = OFF_BEP + (size_t)Bv * Hv * Dv;      // B*H*8*D*D
constexpr size_t OFF_CO   = OFF_SCR + (size_t)Bv * Hv * 8 * Dv * Dv; // B*H*W
constexpr size_t OFF_PW   = OFF_CO + (size_t)Bv * Hv * Wv;       // B*W*W
constexpr size_t OFF_MX   = OFF_PW + (size_t)Bv * Wv * Wv;       // B*W*W

DEVI f32x8 wmma4(f32x2 a, f32x2 b, f32x8 c) {
  // D = A(16x4,f32) * B(4x16,f32) + C(16x16,f32)
  return __builtin_amdgcn_wmma_f32_16x16x4_f32(false, a, false, b, (short)0, c,
                                               false, false);
}

__device__ float block_reduce_sum(float v) {
  __shared__ float sred[256];
  int t = threadIdx.x;
  sred[t] = v;
  __syncthreads();
  for (int s = 128; s > 0; s >>= 1) {
    if (t < s) sred[t] += sred[t + s];
    __syncthreads();
  }
  float r = sred[0];
  __syncthreads();
  return r;
}

DEVI void gsync() { __threadfence(); __syncthreads(); }

// C(64x64) = A(64xK) * B(Kx64) (+ bias per column). K % 64 == 0, blockDim == 256.
// ta: A[m][k] = a[k*lda + m] else a[m*lda + k]
// tb: B[k][n] = b[n*ldb + k] else b[k*ldb + n]
__device__ void gemm64(const float* a, int lda, bool ta,
                       const float* b, int ldb, bool tb,
                       int K, float* out, int ldo, const float* bias) {
  __shared__ float sA[64 * 68];
  __shared__ float sB[64 * 68];
  const int tid = threadIdx.x;
  const int lane = tid & 31;
  const int wave = tid >> 5;
  const int m16 = lane & 15;
  const int hi = lane >> 4;
  const int koff = hi << 1;               // A/B lane k-pair: 0 or 2
  const int si0 = wave >> 2, sj = wave & 3;
  const int si1 = si0 + 2;                // subtile wave+8

  f32x8 acc0 = {0.f, 0.f, 0.f, 0.f, 0.f, 0.f, 0.f, 0.f};
  f32x8 acc1 = {0.f, 0.f, 0.f, 0.f, 0.f, 0.f, 0.f, 0.f};

  for (int k0 = 0; k0 < K; k0 += 64) {
    __syncthreads();
    for (int idx = tid; idx < 64 * 64; idx += 256) {
      int m, kk;
      if (ta) { m = idx & 63; kk = idx >> 6; }   // coalesced over m
      else    { kk = idx & 63; m = idx >> 6; }   // coalesced over k
      sA[m * 68 + kk] = ta ? a[(k0 + kk) * lda + m] : a[m * lda + (k0 + kk)];
      int n, kk2;
      if (tb) { kk2 = idx & 63; n = idx >> 6; }
      else    { n = idx & 63; kk2 = idx >> 6; }
      sB[kk2 * 68 + n] = tb ? b[n * ldb + (k0 + kk2)] : b[(k0 + kk2) * ldb + n];
    }
    __syncthreads();
#pragma unroll
    for (int kl = 0; kl < 64; kl += 4) {
      f32x2 a0, a1, b0;
      a0.x = sA[(si0 * 16 + m16) * 68 + kl + koff];
      a0.y = sA[(si0 * 16 + m16) * 68 + kl + koff + 1];
      a1.x = sA[(si1 * 16 + m16) * 68 + kl + koff];
      a1.y = sA[(si1 * 16 + m16) * 68 + kl + koff + 1];
      b0.x = sB[(kl + koff) * 68 + sj * 16 + m16];
      b0.y = sB[(kl + koff + 1) * 68 + sj * 16 + m16];
      acc0 = wmma4(a0, b0, acc0);   // paired subtiles share B fragment
      acc1 = wmma4(a1, b0, acc1);
    }
  }
  __syncthreads();
  const int col = sj * 16 + m16;
  const float badd = bias ? bias[col] : 0.0f;
#pragma unroll
  for (int v = 0; v < 8; ++v) {
    out[(si0 * 16 + v + hi * 8) * ldo + col] = acc0[v] + badd;
    out[(si1 * 16 + v + hi * 8) * ldo + col] = acc1[v] + badd;
  }
  __syncthreads();
}

// ---------------- Phase kernels ----------------

// K_b = x_b^T x_b  (per-b 512x512, 64x64 tile per block)
__global__ __launch_bounds__(256) void k_xtx(const float* __restrict__ x,
                                             float* __restrict__ ws) {
  int bb = blockIdx.x >> 6;
  int t = blockIdx.x & 63;
  int i0 = (t >> 3) * 64, j0 = (t & 7) * 64;
  const float* xb = x + (size_t)bb * Nv * Wv;
  float* Kb = ws + OFF_K + (size_t)bb * Wv * Wv;
  gemm64(xb + i0, Wv, true, xb + j0, Wv, false, Nv, Kb + i0 * Wv + j0, Wv, nullptr);
}

// xbar_b = column sums of x_b
__global__ __launch_bounds__(256) void k_xbar(const float* __restrict__ x,
                                              float* __restrict__ ws) {
  int bb = blockIdx.x >> 3, g = blockIdx.x & 7;
  int t = threadIdx.x, c = t & 63, part = t >> 6;
  const float* xb = x + (size_t)bb * Nv * Wv;
  float s = 0.f;
  for (int n = part; n < Nv; n += 4) s += xb[(size_t)n * Wv + g * 64 + c];
  __shared__ float red[256];
  red[t] = s;
  __syncthreads();
  if (t < 64)
    ws[OFF_XBAR + bb * Wv + g * 64 + t] = red[t] + red[64 + t] + red[128 + t] + red[192 + t];
}

// trK_b = trace(K_b)
__global__ __launch_bounds__(256) void k_trk(float* __restrict__ ws) {
  int bb = blockIdx.x;
  float s = 0.f;
  for (int w = threadIdx.x; w < Wv; w += 256)
    s += ws[OFF_K + (size_t)bb * Wv * Wv + (size_t)w * Wv + w];
  s = block_reduce_sum(s);
  if (threadIdx.x == 0) ws[OFF_TRK + bb] = s;
}

// Gg_h = Wg Wg^T, gproj_h = Wg g_bias
__global__ __launch_bounds__(256) void k_gg(const float* __restrict__ Wg,
                                            const float* __restrict__ gb,
                                            float* __restrict__ ws) {
  int h = blockIdx.x;
  const float* wgh = Wg + h * Dv * Wv;
  gemm64(wgh, Wv, false, wgh, Wv, true, Wv, ws + OFF_GG + h * Dv * Dv, Dv, nullptr);
  int t = threadIdx.x;
  if (t < 64) {
    float s = 0.f;
    for (int w = 0; w < Wv; ++w) s += wgh[t * Wv + w] * gb[w];
    ws[OFF_GP + h * Dv + t] = s;
  }
}

// KW[bh][0] = K_b @ Wphi_h (512x64),  KW[bh][1] = K_b @ Wg_h^T (512x64)
__global__ __launch_bounds__(256) void k_kw(const float* __restrict__ Wphi,
                                            const float* __restrict__ Wg,
                                            float* __restrict__ ws) {
  int id = blockIdx.x;           // bh*16 + which*8 + itile
  int it = id & 7;
  int which = (id >> 3) & 1;
  int bh = id >> 4;
  int bb = bh >> 3, h = bh & 7;
  int i0 = it * 64;
  const float* Kb = ws + OFF_K + (size_t)bb * Wv * Wv;
  float* outp = ws + OFF_KW + (size_t)bh * 2 * Wv * Dv + (size_t)which * Wv * Dv + i0 * Dv;
  if (which == 0)
    gemm64(Kb + i0 * Wv, Wv, false, Wphi + h * Wv * Dv, Dv, false, Wv, outp, Dv, nullptr);
  else
    gemm64(Kb + i0 * Wv, Wv, false, Wg + h * Dv * Wv, Wv, true, Wv, outp, Dv, nullptr);
}

// Per (b,h): analytic inner step -> We', be', loss0, loss_final, cohead
__global__ __launch_bounds__(256) void k_fast(
    const float* __restrict__ Wphi, const float* __restrict__ bphi,
    const float* __restrict__ We, const float* __restrict__ be,
    const float* __restrict__ Wg, const float* __restrict__ gb,
    const float* __restrict__ bpsi, const float* __restrict__ Wh,
    float* __restrict__ ws, float* __restrict__ dout) {
  const int bh = blockIdx.x;
  const int bb = bh >> 3, h = bh & 7;
  const int tid = threadIdx.x;

  const float* wph = Wphi + h * Wv * Dv;  // [W][D]
  const float* bph = bphi + h * Dv;
  const float* weh = We + h * Dv * Dv;    // [D][D]
  const float* beh = be + h * Dv;
  const float* wgh = Wg + h * Dv * Wv;    // [D][W]
  const float* bps = bpsi + h * Dv;
  const float* whh = Wh + h * Dv * Wv;    // [D][W]
  const float* ggh = ws + OFF_GG + h * Dv * Dv;
  const float* gph = ws + OFF_GP + h * Dv;
  const float* xbar = ws + OFF_XBAR + bb * Wv;
  const float trK = ws[OFF_TRK + bb];
  const float* KWf = ws + OFF_KW + (size_t)bh * 2 * Wv * Dv;
  const float* KWg = KWf + Wv * Dv;
  float* scr = ws + OFF_SCR + (size_t)bh * 8 * Dv * Dv;
  float* MM1 = scr;               float* MN1 = scr + 4096;
  float* MC0 = scr + 2 * 4096;    float* MS = scr + 3 * 4096;
  float* MT = scr + 4 * 4096;     float* MG = scr + 5 * 4096;
  float* MF = scr + 6 * 4096;     float* ME = scr + 7 * 4096;
  float* wep = ws + OFF_WEP + (size_t)bh * Dv * Dv;
  float* bep = ws + OFF_BEP + (size_t)bh * Dv;
  float* coh = ws + OFF_CO + (size_t)bh * Wv;

  __shared__ float svec[8 * 64];
  float* su = svec;        float* sv = svec + 64;
  float* szb = svec + 128; float* sq = svec + 192;
  float* sp = svec + 256;  float* sr = svec + 320;
  float* sp2 = svec + 384; float* sb2 = svec + 448;

  // u = Wphi^T xbar ; v = Wg xbar
  if (tid < 64) {
    float s = 0.f;
    for (int w = 0; w < Wv; ++w) s += wph[w * Dv + tid] * xbar[w];
    su[tid] = s;
  } else if (tid < 128) {
    int e = tid - 64; float s = 0.f;
    for (int w = 0; w < Wv; ++w) s += wgh[e * Wv + w] * xbar[w];
    sv[e] = s;
  }
  __syncthreads();
  if (tid < 64) szb[tid] = su[tid] + Nf * bph[tid];  // zbar = u + N*bphi
  __syncthreads();
  // q = We^T zbar + N*be ; p = We^T bphi + be
  if (tid < 64) {
    float s = 0.f;
    for (int d = 0; d < Dv; ++d) s += weh[d * Dv + tid] * szb[d];
    sq[tid] = s + Nf * beh[tid];
  } else if (tid < 128) {
    int e = tid - 64; float s = 0.f;
    for (int d = 0; d < Dv; ++d) s += weh[d * Dv + e] * bph[d];
    sp[e] = s + beh[e];
  }
  __syncthreads();

  gemm64(wph, Dv, true, KWf, Dv, false, Wv, MM1, Dv, nullptr);  // M1 = Wphi^T K Wphi
  gemm64(wgh, Wv, false, KWf, Dv, false, Wv, MN1, Dv, nullptr); // N1 = Wg K Wphi
  gemm64(wph, Dv, true, KWg, Dv, false, Wv, MC0, Dv, nullptr);  // C0 = Wphi^T K Wg^T
  gsync();
  for (int idx = tid; idx < 4096; idx += 256) {
    int r = idx >> 6, c = idx & 63;
    MC0[idx] += bph[r] * sv[c];
    MS[idx] = MM1[idx] + su[r] * bph[c] + bph[r] * su[c] + Nf * bph[r] * bph[c];
  }
  gsync();
  gemm64(MS, Dv, false, weh, Dv, false, Dv, MT, Dv, nullptr);   // T = S We
  gsync();
  for (int idx = tid; idx < 4096; idx += 256) {
    int r = idx >> 6, c = idx & 63;
    MT[idx] += szb[r] * beh[c];                                  // + zbar be^T
  }
  gsync();
  gemm64(MT, Dv, false, ggh, Dv, false, Dv, MG, Dv, nullptr);   // G = T Gg
  gsync();
  // We' = We - scale*(G + zbar gproj^T - C0)
  for (int idx = tid; idx < 4096; idx += 256) {
    int r = idx >> 6, c = idx & 63;
    wep[idx] = weh[idx] - SCALE * (MG[idx] + szb[r] * gph[c] - MC0[idx]);
  }
  // be' = be - scale*(Gg q + N*gproj - v)
  if (tid < 64) {
    float s = 0.f;
    for (int f = 0; f < Dv; ++f) s += ggh[tid * Dv + f] * sq[f];
    float b2 = beh[tid] - SCALE * (s + Nf * gph[tid] - sv[tid]);
    sb2[tid] = b2;
    bep[tid] = b2;
  }
  gsync();

  // ---- loss0 : 0.5*scale*( tr(We^T M1 We Gg) - 2 tr(We N1) + trK + 2 c^T(A^T xbar) + N c^T c )
  gemm64(MM1, Dv, false, weh, Dv, false, Dv, MF, Dv, nullptr);
  gsync();
  gemm64(weh, Dv, true, MF, Dv, false, Dv, ME, Dv, nullptr);
  gsync();
  float pa = 0.f, pb = 0.f;
  for (int idx = tid; idx < 4096; idx += 256) {
    int r = idx >> 6, c = idx & 63;
    pa += ME[idx] * ggh[idx];
    pb += weh[idx] * MN1[c * 64 + r];
  }
  float t_a = block_reduce_sum(pa);
  float t_b = block_reduce_sum(pb);
  if (tid < 64) {
    float s = 0.f;
    for (int d = 0; d < Dv; ++d) s += weh[d * Dv + tid] * su[d];  // r = We^T u
    sr[tid] = s;
  }
  __syncthreads();
  float acc_cc = 0.f, acc_cv = 0.f;
  for (int w = tid; w < Wv; w += 256) {
    float cw = gb[w], vt = 0.f;
    for (int e = 0; e < Dv; ++e) {
      float g = wgh[e * Wv + w];
      cw += g * sp[e];   // c = Wg^T p + gb
      vt += g * sr[e];   // V^T u = Wg^T r
    }
    acc_cc += cw * cw;
    acc_cv += cw * (vt - xbar[w]);
  }
  float cc = block_reduce_sum(acc_cc);
  float cv = block_reduce_sum(acc_cv);
  if (tid == 0)
    dout[(size_t)Bv * Nv * Wv + bh] =
        0.5f * SCALE * (t_a - 2.0f * t_b + trK + 2.0f * cv + Nf * cc);
  __syncthreads();

  // ---- loss_final: same with We', be'
  gemm64(MM1, Dv, false, wep, Dv, false, Dv, MF, Dv, nullptr);
  gsync();
  gemm64(wep, Dv, true, MF, Dv, false, Dv, ME, Dv, nullptr);
  gsync();
  pa = 0.f; pb = 0.f;
  for (int idx = tid; idx < 4096; idx += 256) {
    int r = idx >> 6, c = idx & 63;
    pa += ME[idx] * ggh[idx];
    pb += wep[idx] * MN1[c * 64 + r];
  }
  t_a = block_reduce_sum(pa);
  t_b = block_reduce_sum(pb);
  if (tid < 64) {
    float s = 0.f, s2 = 0.f;
    for (int d = 0; d < Dv; ++d) {
      float w2 = wep[d * Dv + tid];
      s += w2 * su[d];
      s2 += w2 * bph[d];
    }
    sr[tid] = s;
    sp2[tid] = s2 + sb2[tid];  // c2 inner vector
  }
  __syncthreads();
  acc_cc = 0.f; acc_cv = 0.f;
  for (int w = tid; w < Wv; w += 256) {
    float cw = gb[w], vt = 0.f;
    for (int e = 0; e < Dv; ++e) {
      float g = wgh[e * Wv + w];
      cw += g * sp2[e];
      vt += g * sr[e];
    }
    acc_cc += cw * cw;
    acc_cv += cw * (vt - xbar[w]);
  }
  cc = block_reduce_sum(acc_cc);
  cv = block_reduce_sum(acc_cv);
  if (tid == 0)
    dout[(size_t)Bv * Nv * Wv + 64 + bh] =
        0.5f * SCALE * (t_a - 2.0f * t_b + trK + 2.0f * cv + Nf * cc);
  __syncthreads();

  // ---- cohead = Wh^T (We'^T bpsi + be')
  if (tid < 64) {
    float s = 0.f;
    for (int d = 0; d < Dv; ++d) s += wep[d * Dv + tid] * bps[d];
    sp2[tid] = s + sb2[tid];
  }
  __syncthreads();
  for (int w = tid; w < Wv; w += 256) {
    float s = 0.f;
    for (int e = 0; e < Dv; ++e) s += whh[e * Wv + w] * sp2[e];
    coh[w] = s;
  }
}

// PWcat[b][:, h*64:(h+1)*64] = Wpsi_h @ We'_{b,h}
__global__ __launch_bounds__(256) void k_pw(const float* __restrict__ Wpsi,
                                            float* __restrict__ ws) {
  int id = blockIdx.x;   // bh*8 + itile
  int it = id & 7, bh = id >> 3;
  int bb = bh >> 3, h = bh & 7;
  int i0 = it * 64;
  gemm64(Wpsi + h * Wv * Dv + i0 * Dv, Dv, false,
         ws + OFF_WEP + (size_t)bh * Dv * Dv, Dv, false, Dv,
         ws + OFF_PW + (size_t)bb * Wv * Wv + i0 * Wv + h * Dv, Wv, nullptr);
}

// Mx_b = PWcat[b] @ Wh  (W_h is already a contiguous [512x512] with rows h*64+e)
__global__ __launch_bounds__(256) void k_mx(const float* __restrict__ Wh,
                                            float* __restrict__ ws) {
  int bb = blockIdx.x >> 6;
  int t = blockIdx.x & 63;
  int i0 = (t >> 3) * 64, j0 = (t & 7) * 64;
  gemm64(ws + OFF_PW + (size_t)bb * Wv * Wv + i0 * Wv, Wv, false,
         Wh + j0, Wv, false, Wv,
         ws + OFF_MX + (size_t)bb * Wv * Wv + i0 * Wv + j0, Wv, nullptr);
}

// out_b = x_b @ Mx_b + (sum_h cohead + h_bias)
__global__ __launch_bounds__(256) void k_out(const float* __restrict__ x,
                                             const float* __restrict__ hb,
                                             const float* __restrict__ ws,
                                             float* __restrict__ out) {
  int bb = blockIdx.x >> 9;  // 512 tiles per batch
  int t = blockIdx.x & 511;
  int i0 = (t >> 3) * 64;    // n tile
  int j0 = (t & 7) * 64;     // w tile
  __shared__ float sco[64];
  int tt = threadIdx.x;
  if (tt < 64) {
    float s = hb[j0 + tt];
    for (int h = 0; h < Hv; ++h)
      s += ws[OFF_CO + (size_t)(bb * 8 + h) * Wv + j0 + tt];
    sco[tt] = s;
  }
  __syncthreads();
  gemm64(x + (size_t)bb * Nv * Wv + i0 * Wv, Wv, false,
         ws + OFF_MX + (size_t)bb * Wv * Wv + j0, Wv, false, Wv,
         out + (size_t)bb * Nv * Wv + (size_t)i0 * Wv + j0, Wv, sco);
}

extern "C" void kernel_launch(void* const* d_in, const int* in_sizes, int n_in,
                              void* d_out, int out_size, void* d_ws, size_t ws_size,
                              hipStream_t stream) {
  (void)in_sizes; (void)n_in; (void)out_size; (void)ws_size;
  const float* x    = (const float*)d_in[0];
  const float* Wphi = (const float*)d_in[1];
  const float* bphi = (const float*)d_in[2];
  const float* We   = (const float*)d_in[3];
  const float* be   = (const float*)d_in[4];
  const float* Wg   = (const float*)d_in[5];
  const float* gb   = (const float*)d_in[6];
  const float* Wpsi = (const float*)d_in[7];
  const float* bpsi = (const float*)d_in[8];
  const float* Wh   = (const float*)d_in[9];
  const float* hb   = (const float*)d_in[10];
  float* out = (float*)d_out;
  float* ws = (float*)d_ws;

  k_xtx<<<512, 256, 0, stream>>>(x, ws);
  k_xbar<<<64, 256, 0, stream>>>(x, ws);
  k_trk<<<8, 256, 0, stream>>>(ws);
  k_gg<<<8, 256, 0, stream>>>(Wg, gb, ws);
  k_kw<<<1024, 256, 0, stream>>>(Wphi, Wg, ws);
  k_fast<<<64, 256, 0, stream>>>(Wphi, bphi, We, be, Wg, gb, bpsi, Wh, ws, out);
  k_pw<<<512, 256, 0, stream>>>(Wpsi, ws);
  k_mx<<<512, 256, 0, stream>>>(Wh, ws);
  k_out<<<4096, 256, 0, stream>>>(x, hb, ws, out);
}